// GraphSageConvOD_65944927863278
// MI455X (gfx1250) — compile-verified
//
#include <hip/hip_runtime.h>

#define BB 8
#define NN 2048
#define TT 2
#define BN (BB * NN)
#define DELTA_F 1e-7f

typedef __attribute__((ext_vector_type(2))) float v2f;
typedef __attribute__((ext_vector_type(8))) float v8f;

// ---------------- init: unpack populations + SIR(0:3) ----------------
__global__ void k_init(const float* __restrict__ inf, float* __restrict__ pop,
                       float* __restrict__ SIR) {
  int idx = blockIdx.x * blockDim.x + threadIdx.x;
  if (idx >= BN) return;
  pop[idx] = inf[idx * 4 + 0];
  SIR[idx * 12 + 0] = inf[idx * 4 + 1];
  SIR[idx * 12 + 1] = inf[idx * 4 + 2];
  SIR[idx * 12 + 2] = inf[idx * 4 + 3];
}

__global__ void k_zero(float* __restrict__ p, int n) {
  int idx = blockIdx.x * blockDim.x + threadIdx.x;
  if (idx < n) p[idx] = 0.0f;
}

// ---------------- row pass: row_sum -> scale, invp, OD_out, weights ----------------
// one wave (32 lanes) per row; coalesced streaming read of OD row.
template <int F>
__global__ void k_rowprep(const float* __restrict__ OD, const float* __restrict__ pop,
                          const float* __restrict__ SIR, float* __restrict__ scale_s,
                          float* __restrict__ invp_s, float* __restrict__ odout_s,
                          float* __restrict__ w_s, int t) {
  int gw   = blockIdx.x * 8 + (threadIdx.x >> 5);  // global row id = b*NN + i
  int lane = threadIdx.x & 31;
  int b = gw >> 11;          // NN == 2048
  int i = gw & (NN - 1);
  const float* row = OD + (((size_t)b * TT + t) * NN + i) * NN;
  float s = 0.0f;
  for (int j = lane; j < NN; j += 32) s += row[j];
#pragma unroll
  for (int m = 16; m >= 1; m >>= 1) s += __shfl_xor(s, m, 32);
  if (lane == 0) {
    float p    = pop[gw];
    float den2 = DELTA_F + s;
    float ratio = (den2 == 0.0f) ? 0.0f : p / den2;     // divide_no_nan
    float sc   = (ratio < 1.0f) ? ratio : 1.0f;          // where(ratio<1, OD*ratio, OD)
    float den1 = DELTA_F + p;
    float ivp  = (den1 == 0.0f) ? 0.0f : 1.0f / den1;    // divide_no_nan
    scale_s[gw] = sc;
    invp_s[gw]  = ivp;
    odout_s[gw] = sc * s;                                // OD_out = scale * row_sum
    float swi = sc * ivp;
#pragma unroll
    for (int f = 0; f < F; ++f) w_s[gw * 6 + f] = swi * SIR[gw * 12 + f];
  }
}

// ---------------- column pass: OD_in + SIR_in (rank-F), fused, one OD read ----------------
// grid: (jb = NN/256, ib = NN/256, b). weights staged in LDS; f32 global atomics
// reduce the 8 row-chunk partials.
template <int F>
__global__ void k_colacc(const float* __restrict__ OD, const float* __restrict__ scale_s,
                         const float* __restrict__ w_s, float* __restrict__ odin,
                         float* __restrict__ sirin, int t) {
  __shared__ float sh_s[256];
  __shared__ float sh_w[256 * F];
  int tid = threadIdx.x;
  int b   = blockIdx.z;
  int i0  = blockIdx.y * 256;
  int j   = blockIdx.x * 256 + tid;
  int ib  = b * NN + i0;
  sh_s[tid] = scale_s[ib + tid];
#pragma unroll
  for (int f = 0; f < F; ++f) sh_w[tid * F + f] = w_s[(ib + tid) * 6 + f];
  __syncthreads();

  const float* col = OD + (((size_t)b * TT + t) * NN + i0) * NN + j;
  float a0 = 0.0f;
  float af[F];
#pragma unroll
  for (int f = 0; f < F; ++f) af[f] = 0.0f;
#pragma unroll 4
  for (int ii = 0; ii < 256; ++ii) {
    float v = col[(size_t)ii * NN];
    a0 = fmaf(v, sh_s[ii], a0);
#pragma unroll
    for (int f = 0; f < F; ++f) af[f] = fmaf(v, sh_w[ii * F + f], af[f]);
  }
  atomicAdd(&odin[b * NN + j], a0);
#pragma unroll
  for (int f = 0; f < F; ++f) atomicAdd(&sirin[(b * NN + j) * 6 + f], af[f]);
}

// ---------------- pointwise state update ----------------
template <int F>
__global__ void k_update(float* __restrict__ pop, float* __restrict__ SIR,
                         const float* __restrict__ odout_s, const float* __restrict__ invp_s,
                         const float* __restrict__ odin, const float* __restrict__ sirin) {
  int idx = blockIdx.x * blockDim.x + threadIdx.x;
  if (idx >= BN) return;
  float od_out = odout_s[idx];
  float oin    = odin[idx];
  float ivp    = invp_s[idx];
  float factor = 1.0f - od_out * ivp;  // SIR_stay = SIR*(1 - OD_out*invp)
#pragma unroll
  for (int f = 0; f < F; ++f) SIR[idx * 12 + f] *= factor;
#pragma unroll
  for (int f = 0; f < F; ++f) SIR[idx * 12 + F + f] = sirin[idx * 6 + f];
  pop[idx] += oin - od_out;
}

// ---------------- final projection: (BN x 12) @ (12 x 64) via V_WMMA_F32_16X16X4_F32 ----
// one wave = 16 rows; 3 K-steps of 4 per 16-col tile, 4 tiles -> 12 WMMAs/wave.
__global__ void k_out(const float* __restrict__ SIR, const float* __restrict__ pop,
                      const float* __restrict__ kern, const float* __restrict__ bias,
                      float* __restrict__ out) {
  int wave = blockIdx.x * 8 + (threadIdx.x >> 5);
  int lane = threadIdx.x & 31;
  int half = lane >> 4;   // lanes 16-31 hold K+2 / rows M+8
  int lr   = lane & 15;
  int r0   = wave * 16;
  int row  = r0 + lr;

  // A fragment (16x4 f32): lanes 0-15 -> K={kb,kb+1}; lanes 16-31 -> K={kb+2,kb+3}
  v2f a[3];
#pragma unroll
  for (int kk = 0; kk < 3; ++kk) {
    int kb = kk * 4 + half * 2;
    a[kk].x = SIR[row * 12 + kb];
    a[kk].y = SIR[row * 12 + kb + 1];
  }

#pragma unroll
  for (int ct = 0; ct < 4; ++ct) {
    int col = ct * 16 + lr;
    v8f acc = {0.0f, 0.0f, 0.0f, 0.0f, 0.0f, 0.0f, 0.0f, 0.0f};
#pragma unroll
    for (int kk = 0; kk < 3; ++kk) {
      int kb = kk * 4 + half * 2;
      v2f bf;
      bf.x = kern[kb * 64 + col];
      bf.y = kern[(kb + 1) * 64 + col];
      acc = __builtin_amdgcn_wmma_f32_16x16x4_f32(
          /*neg_a=*/false, a[kk], /*neg_b=*/false, bf,
          /*c_mod=*/(short)0, acc, /*reuse_a=*/false, /*reuse_b=*/false);
    }
    float bs = bias[col];
#pragma unroll
    for (int r = 0; r < 8; ++r) {  // VGPR r holds rows r and r+8
      int orow = r0 + r + half * 8;
      float v = acc[r] + bs;
      v = v > 0.0f ? v : 0.0f;
      out[(size_t)orow * 65 + 1 + col] = v;
    }
  }
  if (half == 0) out[(size_t)row * 65] = pop[row];  // populations -> column 0
}

// ---------------- driver ----------------
extern "C" void kernel_launch(void* const* d_in, const int* in_sizes, int n_in,
                              void* d_out, int out_size, void* d_ws, size_t ws_size,
                              hipStream_t stream) {
  const float* inf  = (const float*)d_in[0];  // (8,2048,4)
  const float* OD   = (const float*)d_in[1];  // (8,2,2048,2048)
  const float* kern = (const float*)d_in[2];  // (12,64)
  const float* bias = (const float*)d_in[3];  // (64,)
  float* out = (float*)d_out;                 // (8,2048,65)

  float* ws      = (float*)d_ws;              // ~1.9 MB of f32 scratch
  float* pop     = ws;                        // BN
  float* SIR     = ws + (size_t)BN;           // BN*12
  float* scale_s = ws + (size_t)BN * 13;      // BN
  float* invp_s  = ws + (size_t)BN * 14;      // BN
  float* odout_s = ws + (size_t)BN * 15;      // BN
  float* w_s     = ws + (size_t)BN * 16;      // BN*6
  float* odin    = ws + (size_t)BN * 22;      // BN   (contiguous with sirin)
  float* sirin   = ws + (size_t)BN * 23;      // BN*6

  k_init<<<BN / 256, 256, 0, stream>>>(inf, pop, SIR);

  // round t=0, F=3
  k_rowprep<3><<<BN / 8, 256, 0, stream>>>(OD, pop, SIR, scale_s, invp_s, odout_s, w_s, 0);
  k_zero<<<(BN * 7 + 255) / 256, 256, 0, stream>>>(odin, BN * 7);
  k_colacc<3><<<dim3(NN / 256, NN / 256, BB), 256, 0, stream>>>(OD, scale_s, w_s, odin, sirin, 0);
  k_update<3><<<BN / 256, 256, 0, stream>>>(pop, SIR, odout_s, invp_s, odin, sirin);

  // round t=1, F=6
  k_rowprep<6><<<BN / 8, 256, 0, stream>>>(OD, pop, SIR, scale_s, invp_s, odout_s, w_s, 1);
  k_zero<<<(BN * 7 + 255) / 256, 256, 0, stream>>>(odin, BN * 7);
  k_colacc<6><<<dim3(NN / 256, NN / 256, BB), 256, 0, stream>>>(OD, scale_s, w_s, odin, sirin, 1);
  k_update<6><<<BN / 256, 256, 0, stream>>>(pop, SIR, odout_s, invp_s, odin, sirin);

  // projection + relu + concat
  k_out<<<BN / 16 / 8, 256, 0, stream>>>(SIR, pop, kern, bias, out);
}